// HCIULayer_85547158602282
// MI455X (gfx1250) — compile-verified
//
#include <hip/hip_runtime.h>
#include <hip/hip_bf16.h>
#include <math.h>

// ---------------- problem constants (match reference) ----------------
static constexpr int B_ = 2, S_ = 1024, H_ = 2048;
static constexpr int KD_ = 32, NCACHE_ = 16;
static constexpr int NTOK_ = B_ * S_;          // 2048 tokens
static constexpr int MAXR_ = 128;
static constexpr float SIM_THRESH_ = 0.95f;
static constexpr float CRIT_T_ = 0.8f, SIMPLE_T_ = 0.3f, EPS_ = 1e-8f;
static constexpr int LDP_ = 68;                // padded LDS row stride (floats): 16B aligned, bank-conflict-free

typedef __attribute__((ext_vector_type(2))) float v2f;
typedef __attribute__((ext_vector_type(8))) float v8f;

// D(16x16,f32) += A(16x4,f32) * B(4x16,f32)   — native CDNA5 f32 matrix op
__device__ __forceinline__ v8f wmma_k4(v2f a, v2f b, v8f c) {
  return __builtin_amdgcn_wmma_f32_16x16x4_f32(
      /*neg_a=*/false, a, /*neg_b=*/false, b,
      /*c_mod=*/(short)0, c, /*reuse_a=*/false, /*reuse_b=*/false);
}

// Fragment loader. For A (16xK row-major): row = lane&15 (=M), cols kbase+2*(lane>=16)+{0,1}.
// For B (KxN where B[k][n] = W[n][k], W row-major): identical pattern with row = lane&15 (=N).
__device__ __forceinline__ v2f load_frag(const float* base, int ld, unsigned lane, int kbase) {
  int row = lane & 15;
  int k = kbase + ((lane >> 4) << 1);
  v2f f;
  f.x = base[(size_t)row * ld + k];
  f.y = base[(size_t)row * ld + k + 1];
  return f;
}

// ---- CDNA5 async DMA: global -> LDS, 16B per lane, tracked by ASYNCcnt ----
__device__ __forceinline__ void async_copy_b128(unsigned lds_off, const float* gptr) {
  asm volatile("global_load_async_to_lds_b128 %0, %1, off"
               :: "v"(lds_off), "v"((unsigned long long)(size_t)gptr)
               : "memory");
}
__device__ __forceinline__ void wait_async0() {
  asm volatile("s_wait_asynccnt 0x0" ::: "memory");
}
// shared-pointer -> wave-relative LDS byte offset (flat LDS addr low 32 bits)
__device__ __forceinline__ unsigned lds_off_of(const void* p) {
  return (unsigned)(size_t)p;
}

// ---------------- kernel 1: x_pooled = mean over S ----------------
__global__ void pool_kernel(const float* __restrict__ x, float* __restrict__ xp) {
  int gid = blockIdx.x * blockDim.x + threadIdx.x;   // 0 .. B*H-1
  if (gid >= B_ * H_) return;
  int b = gid / H_, h = gid - b * H_;
  const float* p = x + (size_t)b * S_ * H_ + h;
  float s = 0.f;
  for (int i = 0; i < S_; ++i) s += p[(size_t)i * H_];
  xp[gid] = s * (1.0f / S_);
}

// ---------------- kernel 2: control (cache 1-NN + rank select) ----------------
__global__ void control_kernel(const float* __restrict__ xp,
                               const float* __restrict__ key_proj_w,
                               const float* __restrict__ cache_keys,
                               const float* __restrict__ ce_w1, const float* __restrict__ ce_b1,
                               const float* __restrict__ ce_w2, const float* __restrict__ ce_b2,
                               int* __restrict__ ctrl) {
  __shared__ float qk[B_ * KD_];
  __shared__ float qfn[B_ * KD_];
  __shared__ float sims[NCACHE_];
  __shared__ float hh[B_ * 64];
  __shared__ float sc[B_ * 4];
  int t = threadIdx.x;
  if (t < B_ * KD_) {
    int b = t / KD_, kd = t - b * KD_;
    const float* xr = xp + (size_t)b * H_;
    const float* kr = key_proj_w + (size_t)kd * H_;
    float s = 0.f;
    for (int h = 0; h < H_; ++h) s += xr[h] * kr[h];
    qk[t] = s;
  }
  __syncthreads();
  if (t < B_ * KD_) {
    int b = t / KD_;
    float nrm = 0.f;
    for (int i = 0; i < KD_; ++i) { float v = qk[b * KD_ + i]; nrm += v * v; }
    qfn[t] = qk[t] / fmaxf(sqrtf(nrm), EPS_);
  }
  __syncthreads();
  if (t < NCACHE_) {
    const float* ckr = cache_keys + (size_t)t * (B_ * KD_);
    float dot = 0.f, cn = 0.f, qn = 0.f;
    for (int i = 0; i < B_ * KD_; ++i) {
      dot += ckr[i] * qfn[i];
      cn += ckr[i] * ckr[i];
      qn += qfn[i] * qfn[i];
    }
    sims[t] = dot / (fmaxf(sqrtf(cn), EPS_) * fmaxf(sqrtf(qn), EPS_));
  }
  if (t < B_ * 64) {
    int b = t >> 6, j = t & 63;
    const float* xr = xp + (size_t)b * H_;
    const float* wr = ce_w1 + (size_t)j * H_;
    float s = ce_b1[j];
    for (int h = 0; h < H_; ++h) s += xr[h] * wr[h];
    hh[t] = fmaxf(s, 0.f);
  }
  __syncthreads();
  if (t < B_ * 4) {
    int b = t >> 2, ri = t & 3;
    float s = ce_b2[ri];
    for (int j = 0; j < 64; ++j) s += hh[b * 64 + j] * ce_w2[ri * 64 + j];
    sc[t] = s;
  }
  __syncthreads();
  if (t == 0) {
    int best = 0; float bs = sims[0];
    for (int i = 1; i < NCACHE_; ++i) if (sims[i] > bs) { bs = sims[i]; best = i; }
    int ridx = 0; float rs = sc[0];
    for (int i = 1; i < B_ * 4; ++i) if (sc[i] > rs) { rs = sc[i]; ridx = i; }
    ridx &= 3;  // flat argmax % len(RANKS)
    const int rv[4] = {4, 12, 40, 128};
    ctrl[0] = best;
    ctrl[1] = (bs >= SIM_THRESH_) ? 1 : 0;
    ctrl[2] = ridx;
    ctrl[3] = rv[ridx];
  }
}

// ---------------- kernel 3: token importance (WMMA scorer) ----------------
// block = 256 (8 waves), each block = 16 tokens; wave w owns feature tiles [w*64, w*64+64)
// x tile staged into LDS via async DMA (ASYNCcnt), padded stride (no bank conflicts).
__global__ void scorer_kernel(const float* __restrict__ x,
                              const float* __restrict__ w1, const float* __restrict__ b1,
                              const float* __restrict__ w2, const float* __restrict__ b2,
                              const float* __restrict__ pos, float* __restrict__ imp) {
  __shared__ float xs[16][LDP_];
  __shared__ float cont[16];
  int rowTile = blockIdx.x;            // 0..127
  int t = threadIdx.x;
  unsigned lane = t & 31;
  int wave = t >> 5;
  if (t < 16) cont[t] = 0.f;
  v8f acc[4] = {{}, {}, {}, {}};
  unsigned xsb = lds_off_of(&xs[0][0]);
  int cr = t >> 4, cc = (t & 15) * 4;  // this thread's 16B slot of the 16x64 tile
  for (int kc = 0; kc < H_; kc += 64) {
    async_copy_b128(xsb + (unsigned)(cr * LDP_ + cc) * 4,
                    x + (size_t)(rowTile * 16 + cr) * H_ + kc + cc);
    wait_async0();
    __syncthreads();
    for (int k4 = 0; k4 < 64; k4 += 4) {
      v2f a = load_frag(&xs[0][0], LDP_, lane, k4);
#pragma unroll
      for (int i = 0; i < 4; ++i) {
        int ct = wave * 4 + i;
        v2f b = load_frag(w1 + (size_t)(ct * 16) * H_ + kc, H_, lane, k4);
        acc[i] = wmma_k4(a, b, acc[i]);
      }
    }
    __syncthreads();
  }
  // fused bias + ReLU + dot with scorer_w2 (ds_add_f32 reduction)
  int n = lane & 15;
  int mo = (lane >> 4) << 3;
#pragma unroll
  for (int i = 0; i < 4; ++i) {
    int f = (wave * 4 + i) * 16 + n;
    float bi = b1[f], wv = w2[f];
#pragma unroll
    for (int r = 0; r < 8; ++r) {
      float hv = fmaxf(acc[i][r] + bi, 0.f);
      atomicAdd(&cont[r + mo], hv * wv);
    }
  }
  __syncthreads();
  if (t < 16) {
    int s = rowTile * 16 + t;
    int srow = s & (S_ - 1);
    float z = cont[t] + b2[0] + 0.1f * pos[srow];
    float im = 1.f / (1.f + expf(-z));
    if (srow == 0 || srow == S_ - 1) im *= 2.f;
    imp[s] = im;
  }
}

// ---------------- kernel 4: P = x @ U_r^T for the device-selected rank ----------------
__global__ void plow_kernel(const float* __restrict__ x,
                            const float* __restrict__ u4, const float* __restrict__ u12,
                            const float* __restrict__ u40, const float* __restrict__ u128,
                            const int* __restrict__ ctrl, float* __restrict__ P) {
  int ridx = ctrl[2];
  int rank = ctrl[3];
  const float* u = (ridx == 0) ? u4 : (ridx == 1) ? u12 : (ridx == 2) ? u40 : u128;
  int ct = blockIdx.x;                  // 0..7  (column tiles of 16)
  int rowTile = blockIdx.y;             // 0..127
  int jb = ct * 16;
  if (jb >= rank) return;               // block-uniform exit (EXEC stays all-ones for WMMA)
  unsigned lane = threadIdx.x & 31;
  int n = lane & 15;
  int jn = jb + n; if (jn >= rank) jn = rank - 1;   // clamp OOB U rows (duplicates, not written)
  const float* urow = u + (size_t)jn * H_;
  const float* xrow = x + (size_t)(rowTile * 16) * H_;
  v8f acc = {};
  for (int k4 = 0; k4 < H_; k4 += 4) {
    v2f a = load_frag(xrow, H_, lane, k4);
    int k = k4 + ((lane >> 4) << 1);
    v2f b; b.x = urow[k]; b.y = urow[k + 1];
    acc = wmma_k4(a, b, acc);
  }
  int mo = (lane >> 4) << 3;
#pragma unroll
  for (int r = 0; r < 8; ++r) {
    int s = rowTile * 16 + r + mo;
    int j = jb + n;
    if (j < rank) P[(size_t)s * MAXR_ + j] = acc[r];
  }
}

// ---------------- kernel 5: P4 = x @ u4^T (rank 4, tiny) ----------------
__global__ void p4_kernel(const float* __restrict__ x, const float* __restrict__ u4,
                          float* __restrict__ P4) {
  int gid = blockIdx.x * blockDim.x + threadIdx.x;  // 0 .. NTOK*4-1
  if (gid >= NTOK_ * 4) return;
  int s = gid >> 2, j = gid & 3;
  const float* xr = x + (size_t)s * H_;
  const float* ur = u4 + (size_t)j * H_;
  float acc = 0.f;
  for (int h = 0; h < H_; ++h) acc += xr[h] * ur[h];
  P4[gid] = acc;
}

// ---------------- kernel 6: fused routed output ----------------
// grid (H/64, NTOK/16), block 128 (4 waves); wave w -> 16 output columns.
// Critical-path GEMM: double-buffered async-DMA staging of both the masked x tile
// (16x64) and the layer_w tile (64x64) in LDS; DMA of chunk k+1 overlaps WMMAs of chunk k.
__global__ void fused_out_kernel(const float* __restrict__ x, const float* __restrict__ imp,
                                 const int* __restrict__ ctrl,
                                 const float* __restrict__ layer_w, const float* __restrict__ layer_b,
                                 const float* __restrict__ cache_deltas,
                                 const float* __restrict__ P, const float* __restrict__ P4,
                                 const float* __restrict__ v4, const float* __restrict__ v12,
                                 const float* __restrict__ v40, const float* __restrict__ v128,
                                 float* __restrict__ out) {
  __shared__ float xs[2][16][LDP_];
  __shared__ float wsh[2][64][LDP_];
  __shared__ float imps[16];
  __shared__ int sAnyCrit;
  int t = threadIdx.x;
  unsigned lane = t & 31;
  int wave = t >> 5;                    // 0..3
  int rowTile = blockIdx.y;
  int nbBlock = blockIdx.x * 64;        // first output column of this block
  int nb = nbBlock + wave * 16;         // first output column of this wave
  if (t == 0) sAnyCrit = 0;
  __syncthreads();
  if (t < 16) {
    float im = imp[rowTile * 16 + t];
    imps[t] = im;
    if (im > CRIT_T_) atomicOr(&sAnyCrit, 1);
  }
  __syncthreads();
  int anyCrit = sAnyCrit;               // block-uniform
  int best = ctrl[0], hit = ctrl[1], ridx = ctrl[2], rank = ctrl[3];
  const float* v = (ridx == 0) ? v4 : (ridx == 1) ? v12 : (ridx == 2) ? v40 : v128;

  // ---- critical path: (masked x) @ layer_w^T, skipped if tile has no critical token ----
  v8f accC = {};
  if (anyCrit) {
    const float* xrow0 = x + (size_t)(rowTile * 16) * H_;
    float amask = (imps[lane & 15] > CRIT_T_) ? 1.f : 0.f;  // per-lane A-row mask (row = lane&15)
    // stage chunk kc into buffer p (all 128 threads issue coalesced 16B DMAs)
    auto stage = [&](int kc, int p) {
      unsigned xb = lds_off_of(&xs[p][0][0]);
      unsigned wb = lds_off_of(&wsh[p][0][0]);
#pragma unroll
      for (int i = 0; i < 2; ++i) {           // 16x64 x-tile: 256 slots of 4 floats
        int slot = t + i * 128;
        int r = slot >> 4, c = (slot & 15) * 4;
        async_copy_b128(xb + (unsigned)(r * LDP_ + c) * 4, xrow0 + (size_t)r * H_ + kc + c);
      }
#pragma unroll
      for (int i = 0; i < 8; ++i) {           // 64x64 W-tile: 1024 slots of 4 floats
        int slot = t + i * 128;
        int r = slot >> 4, c = (slot & 15) * 4;
        async_copy_b128(wb + (unsigned)(r * LDP_ + c) * 4,
                        layer_w + (size_t)(nbBlock + r) * H_ + kc + c);
      }
    };
    stage(0, 0);
    int p = 0;
    for (int kc = 0; kc < H_; kc += 64) {
      wait_async0();        // my DMAs for buffer p have landed
      __syncthreads();      // everyone's DMAs landed; everyone done reading buffer p^1
      if (kc + 64 < H_) stage(kc + 64, p ^ 1);   // overlap next chunk's DMA with compute
      for (int k4 = 0; k4 < 64; k4 += 4) {
        v2f a = load_frag(&xs[p][0][0], LDP_, lane, k4);
        a.x *= amask; a.y *= amask;             // mask non-critical token rows
        v2f b = load_frag(&wsh[p][wave * 16][0], LDP_, lane, k4);
        accC = wmma_k4(a, b, accC);
      }
      p ^= 1;
    }
  }

  // ---- normal path: P(16 x rank) @ V_r^T tile ----
  v8f accN = {};
  {
    const float* prow = P + (size_t)(rowTile * 16) * MAXR_;
    __builtin_prefetch(prow, 0, 1);             // global_prefetch_b8
    int n = lane & 15;
    for (int k4 = 0; k4 < rank; k4 += 4) {
      v2f a = load_frag(prow, MAXR_, lane, k4);
      int k = k4 + ((lane >> 4) << 1);
      v2f b;
      b.x = v[(size_t)(nb + n) * rank + k];
      b.y = v[(size_t)(nb + n) * rank + k + 1];
      accN = wmma_k4(a, b, accN);
    }
  }

  // ---- simple-miss path: P4(16x4) @ v4^T tile (single WMMA) ----
  v8f accS = {};
  {
    v2f a = load_frag(P4 + (size_t)(rowTile * 16) * 4, 4, lane, 0);
    int n = lane & 15;
    int k = (lane >> 4) << 1;
    v2f b;
    b.x = v4[(size_t)(nb + n) * 4 + k];
    b.y = v4[(size_t)(nb + n) * 4 + k + 1];
    accS = wmma_k4(a, b, accS);
  }

  // ---- routed writeback ----
  int n = lane & 15, mo = (lane >> 4) << 3;
  int hcol = nb + n;
  float lb = layer_b[hcol];
#pragma unroll
  for (int r = 0; r < 8; ++r) {
    int m = r + mo;
    int s = rowTile * 16 + m;
    float im = imps[m];
    size_t xi = (size_t)s * H_ + hcol;
    float o;
    if (im > CRIT_T_) {
      o = accC[r] + lb;
    } else {
      float xv = x[xi];
      if (im < SIMPLE_T_) {
        if (hit) {
          int bb = s >> 10;                 // S == 1024
          int srow = s & (S_ - 1);
          o = xv + cache_deltas[(((size_t)best * B_ + bb) * S_ + srow) * H_ + hcol];
        } else {
          o = xv + accS[r];
        }
      } else {
        o = xv + accN[r];
      }
    }
    out[xi] = o;
  }
}

// ---------------- host launcher ----------------
extern "C" void kernel_launch(void* const* d_in, const int* in_sizes, int n_in,
                              void* d_out, int out_size, void* d_ws, size_t ws_size,
                              hipStream_t stream) {
  (void)in_sizes; (void)n_in; (void)out_size; (void)ws_size;
  const float* x        = (const float*)d_in[0];
  const float* sw1      = (const float*)d_in[1];
  const float* sb1      = (const float*)d_in[2];
  const float* sw2      = (const float*)d_in[3];
  const float* sb2      = (const float*)d_in[4];
  const float* pos      = (const float*)d_in[5];
  const float* kproj    = (const float*)d_in[6];
  const float* ckeys    = (const float*)d_in[7];
  const float* cdeltas  = (const float*)d_in[8];
  const float* ce_w1    = (const float*)d_in[9];
  const float* ce_b1    = (const float*)d_in[10];
  const float* ce_w2    = (const float*)d_in[11];
  const float* ce_b2    = (const float*)d_in[12];
  const float* u4       = (const float*)d_in[13];
  const float* v4       = (const float*)d_in[14];
  const float* u12      = (const float*)d_in[15];
  const float* v12      = (const float*)d_in[16];
  const float* u40      = (const float*)d_in[17];
  const float* v40      = (const float*)d_in[18];
  const float* u128     = (const float*)d_in[19];
  const float* v128     = (const float*)d_in[20];
  const float* layer_w  = (const float*)d_in[21];
  const float* layer_b  = (const float*)d_in[22];
  float* out = (float*)d_out;

  // workspace layout (floats)
  float* ws   = (float*)d_ws;
  float* xp   = ws;                       // B*H          = 4096
  int*   ctrl = (int*)(ws + 4096);        // 4 ints
  float* imp  = ws + 4352;                // NTOK         = 2048
  float* P4   = ws + 8192;                // NTOK*4       = 8192
  float* P    = ws + 16384;               // NTOK*MAXR    = 262144

  pool_kernel<<<(B_ * H_ + 255) / 256, 256, 0, stream>>>(x, xp);
  control_kernel<<<1, 256, 0, stream>>>(xp, kproj, ckeys, ce_w1, ce_b1, ce_w2, ce_b2, ctrl);
  scorer_kernel<<<NTOK_ / 16, 256, 0, stream>>>(x, sw1, sb1, sw2, sb2, pos, imp);
  plow_kernel<<<dim3(MAXR_ / 16, NTOK_ / 16), 32, 0, stream>>>(x, u4, u12, u40, u128, ctrl, P);
  p4_kernel<<<(NTOK_ * 4 + 255) / 256, 256, 0, stream>>>(x, u4, P4);
  fused_out_kernel<<<dim3(H_ / 64, NTOK_ / 16), 128, 0, stream>>>(
      x, imp, ctrl, layer_w, layer_b, cdeltas, P, P4, v4, v12, v40, v128, out);
}